// NonSpikingInput_69509750719020
// MI455X (gfx1250) — compile-verified
//
#include <hip/hip_runtime.h>
#include <math.h>

// ---- problem constants (fixed by the reference) ----
#define N_NEUR 4096
#define T_STEPS 8192
// v_{t+1} = A_COEF * v_t + B_COEF * i_t  (RK2 collapsed)
#define A_COEF 0.625f
#define B_COEF 0.075f
// log2(0.625)
#define L2A (-0.6780719051126377f)

// ---- tiling ----
#define SEG 512            // timesteps per workgroup segment (T/SEG = 16 segments)
#define TAPS 64            // FIR lookback; 0.625^64 ~ 8.5e-14 << fp32 noise
#define WIN (TAPS + 16)    // 80-wide input window per 16-output chunk
#define COLS (SEG + TAPS)  // 576 staged columns
#define LROW 580           // LDS row stride (dwords): mult of 4 (b128 aligned),
                           // 580 % 64 == 4 -> 16 lanes hit 16 distinct bank groups

typedef __attribute__((ext_vector_type(2))) float v2f;
typedef __attribute__((ext_vector_type(8))) float v8f;

__global__ __launch_bounds__(256)
void NonSpiking_fir_wmma(const float* __restrict__ in, float* __restrict__ out) {
    __shared__ float lds[16 * LROW];  // 37,120 B

    const int seg    = blockIdx.x;        // time segment [0, T/SEG)
    const int nb     = blockIdx.y * 16;   // neuron tile base
    const int tstart = seg * SEG - TAPS;  // first staged global timestep (may be <0 for seg 0)

    // ---- stage 16 x 576 input slab into LDS (coalesced float4 loads) ----
    const int vec_per_row = COLS / 4;     // 144
    for (int v = threadIdx.x; v < 16 * vec_per_row; v += 256) {
        int row  = v / vec_per_row;
        int c4   = (v - row * vec_per_row) * 4;
        int gcol = tstart + c4;           // multiple of 4 (tstart % 16 == 0 mod sign)
        float4 val = make_float4(0.f, 0.f, 0.f, 0.f);
        if (gcol >= 0)
            val = *(const float4*)(in + (size_t)(nb + row) * T_STEPS + gcol);
        *(float4*)&lds[row * LROW + c4] = val;
    }
    __syncthreads();

    const int lane = threadIdx.x & 31;
    const int wave = threadIdx.x >> 5;
    const int half = lane >> 4;      // 0: lanes 0-15, 1: lanes 16-31
    const int scol = lane & 15;      // N index (output time within chunk) for B/C/D
    const int m    = lane & 15;      // M index (neuron within tile) for A

    // ---- constant Toeplitz B operand, per 16x16x4 K-chunk k (0..19) ----
    // B layout (4x16 f32): VGPR0 = B[4k+2*half, scol], VGPR1 = B[4k+2*half+1, scol]
    // B[j, s] = weight for window col j feeding output s:  d = s + TAPS - j,
    //           w = B_COEF * A_COEF^d  (d >= 0), else 0 (strictly causal).
    float bw0[20], bw1[20];
#pragma unroll
    for (int k = 0; k < 20; ++k) {
        int j0 = 4 * k + 2 * half;
        int d0 = scol + TAPS - j0;
        int d1 = d0 - 1;
        float w0 = B_COEF * exp2f((float)d0 * L2A);
        float w1 = B_COEF * exp2f((float)d1 * L2A);
        bw0[k] = (d0 >= 0) ? w0 : 0.0f;
        bw1[k] = (d1 >= 0) ? w1 : 0.0f;
    }

    // ---- each wave: 4 chunks of 16 output timesteps ----
#pragma unroll
    for (int c = 0; c < 4; ++c) {
        const int wstart = wave * 64 + c * 16;  // LDS col of 80-wide window start
        v8f acc = {0.f, 0.f, 0.f, 0.f, 0.f, 0.f, 0.f, 0.f};
#pragma unroll
        for (int k = 0; k < 20; ++k) {
            // A layout (16x4 f32): lane m holds K = 4k+2*half, 4k+2*half+1
            const float2 a2 =
                *(const float2*)&lds[m * LROW + wstart + 4 * k + 2 * half];
            v2f A; A.x = a2.x; A.y = a2.y;
            v2f B; B.x = bw0[k]; B.y = bw1[k];
            acc = __builtin_amdgcn_wmma_f32_16x16x4_f32(
                false, A, false, B, (short)0, acc, false, false);
        }
        // C/D layout: VGPR r -> (M = r + 8*half, N = scol)
        const int t0 = seg * SEG + wave * 64 + c * 16;
        float* orow = out + (size_t)(nb + 8 * half) * T_STEPS + t0 + scol;
#pragma unroll
        for (int r = 0; r < 8; ++r)
            orow[(size_t)r * T_STEPS] = acc[r];
    }
}

extern "C" void kernel_launch(void* const* d_in, const int* in_sizes, int n_in,
                              void* d_out, int out_size, void* d_ws, size_t ws_size,
                              hipStream_t stream) {
    (void)in_sizes; (void)n_in; (void)d_ws; (void)ws_size; (void)out_size;
    const float* in = (const float*)d_in[0];
    float* out = (float*)d_out;
    dim3 grid(T_STEPS / SEG, N_NEUR / 16);  // 16 x 256 workgroups
    NonSpiking_fir_wmma<<<grid, dim3(256), 0, stream>>>(in, out);
}